// SpGAT_10050223472985
// MI455X (gfx1250) — compile-verified
//
#include <hip/hip_runtime.h>

typedef __attribute__((ext_vector_type(2))) float v2f;
typedef __attribute__((ext_vector_type(8))) float v8f;

#define GAT_ALPHA 0.2f
#define BM 64
#define BN 64
#define BK 32

// Async copy global -> LDS (per-lane LDS destination), tracked by ASYNCcnt.
__device__ __forceinline__ void async_g2l_b32(unsigned lds_byte_off, const float* gsrc)
{
  asm volatile("global_load_async_to_lds_b32 %0, %1, off"
               :: "v"(lds_byte_off), "v"(gsrc) : "memory");
}
__device__ __forceinline__ void wait_asynccnt0()
{
  asm volatile("s_wait_asynccnt 0x0" ::: "memory");
}

// ---------------------------------------------------------------------------
// C[M x Nc] = A[M x K] * B[K x Nc], fp32, V_WMMA_F32_16X16X4_F32.
// Block: 256 threads = 8 waves, 64x64 C tile. Waves arranged 4 (M) x 2 (N);
// each wave owns two 16x16 tiles (cols +0 and +32) sharing one A fragment.
// K staged through LDS in chunks of 32 via GLOBAL_LOAD_ASYNC_TO_LDS_B32
// (A stored transposed As[k][m] during the async copy).
// Nc, K multiples of 64/32 (512 and 64 here). M guarded via clamp + epilogue.
// ---------------------------------------------------------------------------
__global__ __launch_bounds__(256) void gemm_wmma_f32(
    const float* __restrict__ A, const float* __restrict__ B, float* __restrict__ C,
    int M, int K, int lda, int ldb, int ldc)
{
  __shared__ float As[BK][BM + 1];   // As[k][m]
  __shared__ float Bs[BK][BN + 1];   // Bs[k][n]

  const int tid  = threadIdx.x;
  const int lane = tid & 31;
  const int wave = tid >> 5;
  const int wm   = wave & 3;                 // wave's 16-row tile within BM
  const int wn   = wave >> 2;                // wave's first 16-col tile within BN
  const int m0   = blockIdx.x * BM;
  const int n0   = blockIdx.y * BN;

  const int lrow  = lane & 15;               // M (for A) / N (for B) within tile
  const int khalf = (lane >> 4) << 1;        // K offset: lanes 0-15 -> 0, 16-31 -> 2

  const unsigned asBase = (unsigned)(uintptr_t)&As[0][0];
  const unsigned bsBase = (unsigned)(uintptr_t)&Bs[0][0];

  v8f acc0 = {};
  v8f acc1 = {};

  for (int kk = 0; kk < K; kk += BK) {
    // A tile 64x32: consecutive threads read consecutive k (coalesced);
    // transpose into As[k][m] via the per-lane async LDS destination.
#pragma unroll
    for (int i = tid; i < BM * BK; i += 256) {
      int ki = i & (BK - 1);
      int mi = i >> 5;
      int gm = m0 + mi;
      if (gm >= M) gm = M - 1;     // clamp: rows >= M are computed but never stored
      async_g2l_b32(asBase + (unsigned)((ki * (BM + 1) + mi) * 4),
                    A + (size_t)gm * lda + (kk + ki));
    }
    // B tile 32x64: consecutive threads read consecutive n (coalesced)
#pragma unroll
    for (int i = tid; i < BK * BN; i += 256) {
      int ni = i & (BN - 1);
      int ki = i >> 6;
      async_g2l_b32(bsBase + (unsigned)((ki * (BN + 1) + ni) * 4),
                    B + (size_t)(kk + ki) * ldb + (n0 + ni));
    }
    wait_asynccnt0();
    __syncthreads();

#pragma unroll
    for (int k4 = 0; k4 < BK; k4 += 4) {
      v2f a, b0, b1;
      a.x  = As[k4 + khalf    ][wm * 16 + lrow];
      a.y  = As[k4 + khalf + 1][wm * 16 + lrow];
      b0.x = Bs[k4 + khalf    ][wn * 16 + lrow];
      b0.y = Bs[k4 + khalf + 1][wn * 16 + lrow];
      b1.x = Bs[k4 + khalf    ][wn * 16 + 32 + lrow];
      b1.y = Bs[k4 + khalf + 1][wn * 16 + 32 + lrow];
      acc0 = __builtin_amdgcn_wmma_f32_16x16x4_f32(false, a, false, b0,
                                                   (short)0, acc0, false, false);
      acc1 = __builtin_amdgcn_wmma_f32_16x16x4_f32(false, a, false, b1,
                                                   (short)0, acc1, false, false);
    }
    __syncthreads();
  }

  // C/D layout: VGPR v -> row v (lanes 0-15) or v+8 (lanes 16-31), col = lane&15
  const int colA  = n0 + wn * 16 + (lane & 15);
  const int colB  = colA + 32;
  const int rbase = m0 + wm * 16 + ((lane >> 4) << 3);
  if (m0 + BM <= M) {
#pragma unroll
    for (int v = 0; v < 8; ++v) {
      C[(size_t)(rbase + v) * ldc + colA] = acc0[v];
      C[(size_t)(rbase + v) * ldc + colB] = acc1[v];
    }
  } else {
#pragma unroll
    for (int v = 0; v < 8; ++v) {
      if (rbase + v < M) {
        C[(size_t)(rbase + v) * ldc + colA] = acc0[v];
        C[(size_t)(rbase + v) * ldc + colB] = acc1[v];
      }
    }
  }
}

// W_heads[h][k][f] (8x512x64) -> B1[k][h*64+f] (512x512) so layer-1 is one GEMM
__global__ void repack_wheads(const float* __restrict__ Wh, float* __restrict__ B1)
{
  int idx = blockIdx.x * blockDim.x + threadIdx.x;
  if (idx >= 512 * 512) return;
  int k = idx >> 9;
  int c = idx & 511;
  int h = c >> 6, f = c & 63;
  B1[idx] = Wh[((size_t)h * 512 + k) * 64 + f];
}

// s_src[n,h] = H[n, h*64:(h+1)*64] . a[h][:64];  s_dst uses a[h][64:128].
// One wave32 per node; each lane owns a float2 of the 64-wide head slice.
__global__ void scores_kernel(const float* __restrict__ H, int ld, int nheads,
                              const float* __restrict__ a,
                              float* __restrict__ s_src, float* __restrict__ s_dst, int n)
{
  int gw   = (blockIdx.x * blockDim.x + threadIdx.x) >> 5;
  int lane = threadIdx.x & 31;
  if (gw >= n) return;
  for (int h = 0; h < nheads; ++h) {
    const float2 hv = *(const float2*)&H[(size_t)gw * ld + h * 64 + lane * 2];
    const float* ah = a + h * 128;
    float ps = hv.x * ah[lane * 2]      + hv.y * ah[lane * 2 + 1];
    float pd = hv.x * ah[64 + lane * 2] + hv.y * ah[64 + lane * 2 + 1];
    for (int off = 16; off > 0; off >>= 1) {
      ps += __shfl_down(ps, off, 32);
      pd += __shfl_down(pd, off, 32);
    }
    if (lane == 0) { s_src[gw * nheads + h] = ps; s_dst[gw * nheads + h] = pd; }
  }
}

// One wave32 per edge: w = exp(-leakyrelu(s_src[src]+s_dst[dst])) per head,
// then coalesced gather of H[dst] row + GLOBAL_ATOMIC_ADD_F32 scatter to hp[src].
__global__ void edge_kernel(const int* __restrict__ src, const int* __restrict__ dst,
                            const float* __restrict__ H, int ld, int nheads,
                            const float* __restrict__ s_src, const float* __restrict__ s_dst,
                            float* __restrict__ hp, float* __restrict__ rowsum, int ne)
{
  int gw   = (blockIdx.x * blockDim.x + threadIdx.x) >> 5;
  int lane = threadIdx.x & 31;
  if (gw >= ne) return;
  const int s = src[gw];
  const int d = dst[gw];
  for (int h = 0; h < nheads; ++h) {
    float sc = s_src[s * nheads + h] + s_dst[d * nheads + h];
    float lr = sc > 0.0f ? sc : GAT_ALPHA * sc;
    float w  = __expf(-lr);
    if (lane == 0) atomicAdd(&rowsum[s * nheads + h], w);
    int f = h * 64 + lane * 2;
    const float2 hv = *(const float2*)&H[(size_t)d * ld + f];
    atomicAdd(&hp[(size_t)s * ld + f],     w * hv.x);
    atomicAdd(&hp[(size_t)s * ld + f + 1], w * hv.y);
  }
}

// out[n][c] = elu(hp[n][c] / guarded rowsum[n][c/64]); in-place safe.
__global__ void norm_elu_kernel(const float* __restrict__ hp, const float* __restrict__ rowsum,
                                int nheads, float* __restrict__ out, int n)
{
  const int width = nheads * 64;
  long long idx = (long long)blockIdx.x * blockDim.x + threadIdx.x;
  if (idx >= (long long)n * width) return;
  int c   = (int)(idx % width);
  int row = (int)(idx / width);
  float rs = rowsum[row * nheads + (c >> 6)];
  rs = (rs == 0.0f) ? 1.0f : rs;
  float v = hp[idx] / rs;
  out[idx] = v > 0.0f ? v : expm1f(v);
}

extern "C" void kernel_launch(void* const* d_in, const int* in_sizes, int n_in,
                              void* d_out, int out_size, void* d_ws, size_t ws_size,
                              hipStream_t stream)
{
  const float* x    = (const float*)d_in[0];   // [N, 512]
  const int*   src  = (const int*)d_in[1];     // [E]
  const int*   dst  = (const int*)d_in[2];     // [E]
  const float* Wh   = (const float*)d_in[3];   // [8, 512, 64]
  const float* ah   = (const float*)d_in[4];   // [8, 128]
  const float* Wout = (const float*)d_in[5];   // [512, 64]
  const float* aout = (const float*)d_in[6];   // [128]

  const int N = in_sizes[0] / 512;
  const int E = in_sizes[1];

  // workspace carve (fp32): ~237 MB total
  float* p   = (float*)d_ws;
  float* B1  = p; p += 512 * 512;          // packed layer-1 weights
  float* H1  = p; p += (size_t)N * 512;    // layer-1 features (all heads)
  float* HP1 = p; p += (size_t)N * 512;    // layer-1 accumulator -> xc
  float* Ss1 = p; p += (size_t)N * 8;
  float* Sd1 = p; p += (size_t)N * 8;
  float* Rs1 = p; p += (size_t)N * 8;
  float* H2  = p; p += (size_t)N * 64;
  float* HP2 = p; p += (size_t)N * 64;
  float* Ss2 = p; p += N;
  float* Sd2 = p; p += N;
  float* Rs2 = p; /* p += N; */

  hipMemsetAsync(HP1, 0, (size_t)N * 512 * sizeof(float), stream);
  hipMemsetAsync(Rs1, 0, (size_t)N * 8 * sizeof(float), stream);
  hipMemsetAsync(HP2, 0, (size_t)N * 64 * sizeof(float), stream);
  hipMemsetAsync(Rs2, 0, (size_t)N * sizeof(float), stream);

  repack_wheads<<<(512 * 512 + 255) / 256, 256, 0, stream>>>(Wh, B1);

  // ---- layer 1 ----
  {
    dim3 g((N + BM - 1) / BM, 512 / BN);
    gemm_wmma_f32<<<g, 256, 0, stream>>>(x, B1, H1, N, 512, 512, 512, 512);
  }
  scores_kernel<<<(N * 32 + 255) / 256, 256, 0, stream>>>(H1, 512, 8, ah, Ss1, Sd1, N);
  edge_kernel<<<(int)(((size_t)E * 32 + 255) / 256), 256, 0, stream>>>(
      src, dst, H1, 512, 8, Ss1, Sd1, HP1, Rs1, E);
  norm_elu_kernel<<<(int)(((size_t)N * 512 + 255) / 256), 256, 0, stream>>>(
      HP1, Rs1, 8, HP1, N);   // HP1 becomes xc in place

  // ---- layer 2 ----
  {
    dim3 g((N + BM - 1) / BM, 64 / BN);
    gemm_wmma_f32<<<g, 256, 0, stream>>>(HP1, Wout, H2, N, 512, 512, 64, 64);
  }
  scores_kernel<<<(N * 32 + 255) / 256, 256, 0, stream>>>(H2, 64, 1, aout, Ss2, Sd2, N);
  edge_kernel<<<(int)(((size_t)E * 32 + 255) / 256), 256, 0, stream>>>(
      src, dst, H2, 64, 1, Ss2, Sd2, HP2, Rs2, E);
  norm_elu_kernel<<<(int)(((size_t)N * 64 + 255) / 256), 256, 0, stream>>>(
      HP2, Rs2, 1, (float*)d_out, N);
}